// NMS3dAndComposeA_22857815949342
// MI455X (gfx1250) — compile-verified
//
#include <hip/hip_runtime.h>
#include <hip/hip_bf16.h>
#include <stdint.h>

// ---------------------------------------------------------------------------
// MI455X (gfx1250) multi-scale NMS + top-k + soft-argmax LAF composition.
// - K1 stages 66x72x3 halo tiles in LDS via CDNA5 async-to-LDS b128 (ASYNCcnt)
// - top-k via 2x16-bit radix select + bitonic sort
// - K6 soft-argmax refinement as a real WMMA GEMM: D[16x16] = R[16x28]*Wt[28x16]
//   (V_WMMA_F32_16X16X4_F32, 7 K-chunks; cols 0..3 = den, z-, y-, x-moments)
// ---------------------------------------------------------------------------

#define NBINS      65536
#define CAND_CAP   (1u << 19)
#define SEL_CAP    4096
#define TIE_CAP    2048
#define TILE       64
#define HROWS      66
#define LROW       72
#define NCHUNK     18
#define PS         (HROWS * LROW)

typedef __attribute__((ext_vector_type(2))) float v2f;
typedef __attribute__((ext_vector_type(8))) float v8f;

__device__ __forceinline__ unsigned okey(float f) {
  unsigned u = __float_as_uint(f);
  return (u & 0x80000000u) ? ~u : (u | 0x80000000u);
}
__device__ __forceinline__ float unokey(unsigned k) {
  return __uint_as_float((k & 0x80000000u) ? (k ^ 0x80000000u) : ~k);
}
__device__ __forceinline__ unsigned umin_d(unsigned a, unsigned b) { return a < b ? a : b; }

__device__ __forceinline__ void async_g2l_b128(unsigned lds_byte, const float* g) {
  unsigned long long ga = (unsigned long long)g;
  asm volatile("global_load_async_to_lds_b128 %0, %1, off"
               :: "v"(lds_byte), "v"(ga) : "memory");
}
__device__ __forceinline__ void wait_async0() {
  asm volatile("s_wait_asynccnt 0" ::: "memory");
}

// ---------------------------------------------------------------------------
__global__ __launch_bounds__(256) void k_init(unsigned* cnt, unsigned* hist1, unsigned* hist2) {
  unsigned i = blockIdx.x * blockDim.x + threadIdx.x;
  unsigned stride = gridDim.x * blockDim.x;
  for (unsigned j = i; j < 16u; j += stride) cnt[j] = 0u;
  for (unsigned j = i; j < (unsigned)NBINS; j += stride) { hist1[j] = 0u; hist2[j] = 0u; }
}

// ---------------------------------------------------------------------------
// K1: tiled 3x3x3 NMS with async-to-LDS staging.
// ---------------------------------------------------------------------------
__global__ __launch_bounds__(256) void k_nms(const float* __restrict__ low,
                                             const float* __restrict__ cur,
                                             const float* __restrict__ high,
                                             int H, int W,
                                             unsigned* cnt, unsigned* hist1,
                                             float* candV, unsigned* candI) {
  extern __shared__ float S[];  // 3 * PS floats = 57024 B
  const int tid = threadIdx.x;
  const int tilesX = (W + TILE - 1) / TILE;
  const int tx = blockIdx.x % tilesX;
  const int ty = blockIdx.x / tilesX;
  const int x0 = tx * TILE, y0 = ty * TILE;

  for (int l = tid; l < HROWS * NCHUNK; l += 256) {
    int r = l / NCHUNK, cc = l % NCHUNK;
    int gy = y0 - 1 + r; gy = gy < 0 ? 0 : (gy >= H ? H - 1 : gy);
    int gxs = x0 - 4 + cc * 4;
    if (gxs < 0 || gxs + 3 >= W) continue;
    unsigned goff = (unsigned)gy * (unsigned)W + (unsigned)gxs;
    unsigned ldsb = (unsigned)((r * LROW + cc * 4) * 4);
    async_g2l_b128(ldsb,              low  + goff);
    async_g2l_b128(ldsb + PS * 4,     cur  + goff);
    async_g2l_b128(ldsb + 2 * PS * 4, high + goff);
  }
  wait_async0();
  __syncthreads();

  for (int i = 0; i < (TILE * TILE) / 256; ++i) {
    int o = i * 256 + tid;
    int r = o >> 6, c = o & (TILE - 1);
    int y = y0 + r, x = x0 + c;
    if (y < 1 || y >= H - 1 || x < 1 || x >= W - 1) continue;
    float v = S[PS + (r + 1) * LROW + (c + 4)];
    float m = -__builtin_inff();
#pragma unroll
    for (int ch = 0; ch < 3; ++ch) {
      int base = ch * PS + r * LROW + (c + 3);
#pragma unroll
      for (int dy = 0; dy < 3; ++dy) {
        int row = base + dy * LROW;
        m = fmaxf(m, S[row]);
        m = fmaxf(m, S[row + 1]);
        m = fmaxf(m, S[row + 2]);
      }
    }
    if (v - m + 1e-5f > 0.0f) {
      unsigned pos = atomicAdd(&cnt[0], 1u);
      if (pos < CAND_CAP) {
        candV[pos] = v;
        candI[pos] = (unsigned)y * (unsigned)W + (unsigned)x;
        atomicAdd(&hist1[okey(v) >> 16], 1u);
      }
    }
  }
}

// ---------------------------------------------------------------------------
// K2/K4: radix-select scans; K3/K5: compaction passes (unchanged).
// ---------------------------------------------------------------------------
__global__ __launch_bounds__(256) void k_scan1(unsigned* cnt, const unsigned* hist1, int k) {
  __shared__ unsigned part[256];
  int tid = threadIdx.x;
  unsigned s = 0;
  for (int b = tid * 256; b < tid * 256 + 256; ++b) s += hist1[b];
  part[tid] = s;
  __syncthreads();
  if (tid == 0) {
    unsigned nc = umin_d(cnt[0], CAND_CAP);
    unsigned keff = umin_d((unsigned)k, nc);
    cnt[8] = keff;
    if (keff == 0u) { cnt[3] = 0x20000u; cnt[4] = 0u; return; }
    unsigned cum = 0; int chunk = 255;
    for (; chunk > 0; --chunk) {
      if (cum + part[chunk] >= keff) break;
      cum += part[chunk];
    }
    unsigned T1 = 0, above = cum;
    for (int b = chunk * 256 + 255; b >= chunk * 256; --b) {
      unsigned h = hist1[b];
      if (above + h >= keff) { T1 = (unsigned)b; break; }
      above += h;
    }
    cnt[3] = T1; cnt[4] = above;
  }
}

__global__ __launch_bounds__(256) void k_pass1(unsigned* cnt,
                                               const float* candV, const unsigned* candI,
                                               unsigned* hist2, float* selV, unsigned* selI) {
  unsigned nc = umin_d(cnt[0], CAND_CAP);
  unsigned T1 = cnt[3];
  unsigned stride = gridDim.x * blockDim.x;
  for (unsigned i = blockIdx.x * blockDim.x + threadIdx.x; i < nc; i += stride) {
    unsigned key = okey(candV[i]);
    unsigned hb = key >> 16;
    if (hb > T1) {
      unsigned p = atomicAdd(&cnt[1], 1u);
      if (p < SEL_CAP) { selV[p] = candV[i]; selI[p] = candI[i]; }
    } else if (hb == T1) {
      atomicAdd(&hist2[key & 0xFFFFu], 1u);
    }
  }
}

__global__ __launch_bounds__(256) void k_scan2(unsigned* cnt, const unsigned* hist2) {
  __shared__ unsigned part[256];
  int tid = threadIdx.x;
  unsigned s = 0;
  for (int b = tid * 256; b < tid * 256 + 256; ++b) s += hist2[b];
  part[tid] = s;
  __syncthreads();
  if (tid == 0) {
    unsigned keff = cnt[8], above1 = cnt[4], T1 = cnt[3];
    if (keff == 0u || T1 > 0xFFFFu) { cnt[5] = 0xFFFFFFFFu; cnt[6] = 0u; cnt[7] = 0u; return; }
    unsigned need = keff - above1;
    unsigned cum = 0; int chunk = 255;
    for (; chunk > 0; --chunk) {
      if (cum + part[chunk] >= need) break;
      cum += part[chunk];
    }
    unsigned T2 = 0, above = cum;
    for (int b = chunk * 256 + 255; b >= chunk * 256; --b) {
      unsigned h = hist2[b];
      if (above + h >= need) { T2 = (unsigned)b; break; }
      above += h;
    }
    cnt[5] = (T1 << 16) | T2;
    cnt[6] = above1 + above;
    cnt[7] = keff - (above1 + above);
  }
}

__global__ __launch_bounds__(256) void k_pass2(unsigned* cnt,
                                               const float* candV, const unsigned* candI,
                                               float* selV, unsigned* selI, unsigned* tieI) {
  unsigned nc = umin_d(cnt[0], CAND_CAP);
  unsigned T1 = cnt[3], Kstar = cnt[5];
  unsigned stride = gridDim.x * blockDim.x;
  for (unsigned i = blockIdx.x * blockDim.x + threadIdx.x; i < nc; i += stride) {
    unsigned key = okey(candV[i]);
    if (key > Kstar && (key >> 16) == T1) {
      unsigned p = atomicAdd(&cnt[1], 1u);
      if (p < SEL_CAP) { selV[p] = candV[i]; selI[p] = candI[i]; }
    } else if (key == Kstar) {
      unsigned p = atomicAdd(&cnt[2], 1u);
      if (p < TIE_CAP) tieI[p] = candI[i];
    }
  }
}

// ---------------------------------------------------------------------------
// K6: bitonic-sort winners/ties, then WMMA soft-argmax + LAF composition.
// ---------------------------------------------------------------------------
__device__ void bitonic_asc(unsigned long long* a, int N) {
  int tid = threadIdx.x, nth = blockDim.x;
  for (int ks = 2; ks <= N; ks <<= 1) {
    for (int j = ks >> 1; j > 0; j >>= 1) {
      for (int i = tid; i < N; i += nth) {
        int ixj = i ^ j;
        if (ixj > i) {
          bool up = ((i & ks) == 0);
          unsigned long long x = a[i], y = a[ixj];
          if ((x > y) == up) { a[i] = y; a[ixj] = x; }
        }
      }
      __syncthreads();
    }
  }
}

// weight matrix Wt[28][16]: cols 0..3 = (den, z, y, x) moments, rest 0
__device__ __forceinline__ float wmat(int j, int n) {
  if (j >= 27 || n > 3) return 0.0f;
  int ch = j / 9, rem = j % 9;
  int dy = rem / 3 - 1, dx = rem % 3 - 1;
  if (n == 0) return 1.0f;
  if (n == 1) return (float)(ch - 1);
  if (n == 2) return (float)dy;
  return (float)dx;
}

// neighborhood element j (0..26) of feature at (y,x); 0 if invalid/padded
__device__ __forceinline__ float nbhd(const float* __restrict__ low,
                                      const float* __restrict__ cur,
                                      const float* __restrict__ high,
                                      int W, int y, int x, int j, bool valid) {
  if (!valid || j >= 27) return 0.0f;
  int ch = j / 9, rem = j % 9;
  int dy = rem / 3 - 1, dx = rem % 3 - 1;
  unsigned o = (unsigned)(y + dy) * (unsigned)W + (unsigned)(x + dx);
  const float* p = (ch == 0) ? low : ((ch == 1) ? cur : high);
  return p[o];
}

__global__ __launch_bounds__(256) void k_final(const unsigned* cnt,
                                               const float* selV, const unsigned* selI,
                                               const unsigned* tieI,
                                               const float* __restrict__ low,
                                               const float* __restrict__ cur,
                                               const float* __restrict__ high,
                                               int H, int W, int k, float* out) {
  __shared__ unsigned long long SA[SEL_CAP];  // 32 KB winners
  __shared__ unsigned long long ST[TIE_CAP];  // 16 KB ties
  __shared__ float DM[8][16][16];             // 8 KB per-wave D spill
  const int tid = threadIdx.x, nth = blockDim.x;
  unsigned nsel = umin_d(umin_d(cnt[1], (unsigned)SEL_CAP), cnt[6]);
  unsigned ntie = umin_d(cnt[2], (unsigned)TIE_CAP);
  unsigned keff = cnt[8];
  unsigned Kstar = cnt[5];
  unsigned tieNeed = umin_d(cnt[7], ntie);

  for (unsigned i = tid; i < SEL_CAP; i += nth)
    SA[i] = (i < nsel) ? ((((unsigned long long)(~okey(selV[i]))) << 32) | (unsigned long long)selI[i])
                       : ~0ull;
  for (unsigned i = tid; i < TIE_CAP; i += nth)
    ST[i] = (i < ntie) ? (unsigned long long)tieI[i] : ~0ull;
  __syncthreads();
  bitonic_asc(SA, SEL_CAP);
  bitonic_asc(ST, TIE_CAP);  // ends with __syncthreads

  const float vstar = unokey(Kstar);
  const float ms = (float)(H < W ? H : W);

  // 16 features per wave-group; WMMA in wave-uniform control flow.
  const int wave = tid >> 5, lane = tid & 31;
  const int h = lane >> 4;       // K-half (A/B layout)
  const int m = lane & 15;       // feature-in-group / N column
  const int ngroups = (k + 15) / 16;

  for (int g = wave; g < ngroups; g += 8) {
    int r = g * 16 + m;
    float val = 0.0f; unsigned idx = 0u; bool valid = false;
    if ((unsigned)r < nsel) {
      unsigned long long e = SA[r];
      idx = (unsigned)(e & 0xFFFFFFFFull);
      val = unokey(~(unsigned)(e >> 32));
      valid = true;
    } else if (r < k && (unsigned)r < keff && (unsigned)(r - (int)nsel) < tieNeed) {
      idx = (unsigned)ST[r - (int)nsel];
      val = vstar;
      valid = true;
    }
    if (h == 0 && r < k) out[r] = val;
    int y = valid ? (int)(idx / (unsigned)W) : 1;
    int x = valid ? (int)(idx % (unsigned)W) : 1;

    // D[16x16] = sum_t A_t(16x4) * B_t(4x16):  R[16x28] x Wt[28x16]
    v8f acc = {};
#pragma unroll
    for (int t = 0; t < 7; ++t) {
      int j0 = 4 * t + 2 * h;
      v2f A, B;
      A.x = nbhd(low, cur, high, W, y, x, j0,     valid);
      A.y = nbhd(low, cur, high, W, y, x, j0 + 1, valid);
      B.x = wmat(j0,     m);
      B.y = wmat(j0 + 1, m);
      acc = __builtin_amdgcn_wmma_f32_16x16x4_f32(false, A, false, B,
                                                  (short)0, acc, false, false);
    }
    // spill D: lane(h,m) VGPR v holds D[M = v + 8h][N = m]
#pragma unroll
    for (int v = 0; v < 8; ++v) DM[wave][v + 8 * h][m] = acc[v];
    asm volatile("s_wait_dscnt 0" ::: "memory");  // same-wave LDS RAW

    if (h == 0 && r < k) {
      float l0v = 0.f, l2v = 0.f, l4v = 0.f, l5v = 0.f;
      if (valid) {
        float den = DM[wave][m][0];
        float nz  = DM[wave][m][1];
        float ny  = DM[wave][m][2];
        float nx  = DM[wave][m][3];
        float inv = 1.0f / (den + 1e-8f);
        float scz = nz * inv;
        float scy = ny * inv + (float)y;
        float scx = nx * inv + (float)x;
        float scale = scz / ms;
        l0v = scale; l2v = scx / (float)W; l4v = scale; l5v = scy / (float)H;
      }
      float* lp = out + k + r * 6;
      lp[0] = l0v; lp[1] = 0.0f; lp[2] = l2v;
      lp[3] = 0.0f; lp[4] = l4v; lp[5] = l5v;
    }
  }
}

// ---------------------------------------------------------------------------
extern "C" void kernel_launch(void* const* d_in, const int* in_sizes, int n_in,
                              void* d_out, int out_size, void* d_ws, size_t ws_size,
                              hipStream_t stream) {
  const float* low  = (const float*)d_in[0];
  const float* cur  = (const float*)d_in[1];
  const float* high = (const float*)d_in[2];
  float* out = (float*)d_out;

  long long n0 = (long long)in_sizes[0];
  int W = 1; while ((long long)W * W < n0) W <<= 1;
  int H = (int)(n0 / W);
  int k = out_size / 7;

  char* ws = (char*)d_ws;
  unsigned* cnt   = (unsigned*)ws;
  unsigned* hist1 = (unsigned*)(ws + 256);
  unsigned* hist2 = (unsigned*)(ws + 256 + NBINS * 4);
  float*    candV = (float*)   (ws + 256 + 2 * NBINS * 4);
  unsigned* candI = (unsigned*)((char*)candV + (size_t)CAND_CAP * 4);
  float*    selV  = (float*)   ((char*)candI + (size_t)CAND_CAP * 4);
  unsigned* selI  = (unsigned*)((char*)selV + SEL_CAP * 4);
  unsigned* tieI  = (unsigned*)((char*)selI + SEL_CAP * 4);
  (void)ws_size; (void)n_in;

  k_init<<<512, 256, 0, stream>>>(cnt, hist1, hist2);

  int tilesX = (W + TILE - 1) / TILE;
  int tilesY = (H + TILE - 1) / TILE;
  size_t shmem = (size_t)(3 * PS) * sizeof(float);  // 57024 B
  k_nms<<<tilesX * tilesY, 256, shmem, stream>>>(low, cur, high, H, W,
                                                 cnt, hist1, candV, candI);

  k_scan1<<<1, 256, 0, stream>>>(cnt, hist1, k);
  k_pass1<<<512, 256, 0, stream>>>(cnt, candV, candI, hist2, selV, selI);
  k_scan2<<<1, 256, 0, stream>>>(cnt, hist2);
  k_pass2<<<512, 256, 0, stream>>>(cnt, candV, candI, selV, selI, tieI);
  k_final<<<1, 256, 0, stream>>>(cnt, selV, selI, tieI, low, cur, high, H, W, k, out);
}